// GCN_5334349382370
// MI455X (gfx1250) — compile-verified
//
#include <hip/hip_runtime.h>

// ---------------------------------------------------------------------------
// Types for CDNA5 WMMA (wave32): fp32 16x16x4 -> A,B = 2 VGPRs, C/D = 8 VGPRs
// ---------------------------------------------------------------------------
typedef __attribute__((ext_vector_type(2))) float v2f;
typedef __attribute__((ext_vector_type(8))) float v8f;

#define DH 256  // feature width (D_IN == D_HID == 256)

// ---------------------------------------------------------------------------
// Degree / normalization:  deg[i] = 1 (self loop) + indegree(i);  dinv = deg^-1/2
// ---------------------------------------------------------------------------
__global__ void gcn_deg_init(float* __restrict__ deg, int n) {
    int i = blockIdx.x * blockDim.x + threadIdx.x;
    if (i < n) deg[i] = 1.0f;   // self-loop contribution
}

__global__ void gcn_deg_count(const int* __restrict__ dst, float* __restrict__ deg, int e) {
    int i = blockIdx.x * blockDim.x + threadIdx.x;
    if (i < e) atomicAdd(deg + dst[i], 1.0f);
}

__global__ void gcn_deg_rsqrt(float* __restrict__ deg, int n) {
    int i = blockIdx.x * blockDim.x + threadIdx.x;
    if (i < n) deg[i] = rsqrtf(deg[i]);   // deg >= 1 always (self loop)
}

// ---------------------------------------------------------------------------
// Dense GEMM via V_WMMA_F32_16X16X4_F32.
// C[M,256] = A[M,256] @ B[256,256], M % 16 == 0.
//
// Block = 128 threads = 4 waves; each wave computes a 16x64 strip of C
// (4 accumulators), so one block covers a full 16x256 output tile and each
// A fragment is loaded once and reused against 4 B fragments (4x less A
// traffic, load:wmma ratio 10B/lane per wmma). grid = (M/16).
//
// Fragment layouts (ISA 7.12.2):
//   A 16x4 f32 : lane<16 -> row=lane,  {K=kk, K=kk+1};  lane>=16 -> {K=kk+2, kk+3}
//   B  4x16 f32: lane<16 -> col=lane,  rows {kk, kk+1}; lane>=16 -> rows {kk+2, kk+3}
//   C 16x16 f32: VGPR v, lane -> M = v + 8*(lane>=16), N = lane&15
// ---------------------------------------------------------------------------
__global__ __launch_bounds__(128) void gcn_gemm_wmma(
    const float* __restrict__ A, const float* __restrict__ B,
    float* __restrict__ C, int M) {
    const int lane = threadIdx.x & 31;
    const int wave = threadIdx.x >> 5;
    const int half = lane >> 4;        // 0 | 1
    const int l15  = lane & 15;
    const int tm   = blockIdx.x;
    const int colb = wave * 64;        // this wave's 64-column strip

    const int row = tm * 16 + l15;

    const float* arow = A + (size_t)row * DH + 2 * half;            // contiguous pair
    const float* bcol = B + (size_t)(2 * half) * DH + colb + l15;   // stride-DH pairs

    v8f acc0 = {}, acc1 = {}, acc2 = {}, acc3 = {};
    #pragma unroll 4
    for (int kk = 0; kk < DH; kk += 4) {
        v2f a = *(const v2f*)(arow + kk);          // global_load_b64, shared by 4 wmma
        const float* bk = bcol + (size_t)kk * DH;
        v2f b0, b1, b2, b3;
        b0.x = bk[0];  b0.y = bk[DH];
        b1.x = bk[16]; b1.y = bk[DH + 16];
        b2.x = bk[32]; b2.y = bk[DH + 32];
        b3.x = bk[48]; b3.y = bk[DH + 48];
        acc0 = __builtin_amdgcn_wmma_f32_16x16x4_f32(false, a, false, b0, (short)0, acc0, false, false);
        acc1 = __builtin_amdgcn_wmma_f32_16x16x4_f32(false, a, false, b1, (short)0, acc1, false, false);
        acc2 = __builtin_amdgcn_wmma_f32_16x16x4_f32(false, a, false, b2, (short)0, acc2, false, false);
        acc3 = __builtin_amdgcn_wmma_f32_16x16x4_f32(false, a, false, b3, (short)0, acc3, false, false);
    }

    float* crow = C + (size_t)(tm * 16 + half * 8) * DH + colb + l15;
    #pragma unroll
    for (int v = 0; v < 8; ++v) {
        crow[(size_t)v * DH +  0] = acc0[v];
        crow[(size_t)v * DH + 16] = acc1[v];
        crow[(size_t)v * DH + 32] = acc2[v];
        crow[(size_t)v * DH + 48] = acc3[v];
    }
}

// ---------------------------------------------------------------------------
// Aggregation init: out[i] = bias + h[i] * dinv[i]^2   (self-loop term)
// ---------------------------------------------------------------------------
__global__ __launch_bounds__(256) void gcn_agg_init(
    const float* __restrict__ h, const float* __restrict__ dinv,
    const float* __restrict__ bias, float* __restrict__ out) {
    int i = blockIdx.x;
    int f = threadIdx.x;
    float di = dinv[i];
    out[(size_t)i * DH + f] = bias[f] + h[(size_t)i * DH + f] * di * di;
}

// ---------------------------------------------------------------------------
// Edge scatter: out[dst] += h[src] * dinv[src]*dinv[dst]
// 4 edges per 256-thread block; 64 lanes/edge; float4 gather + 4 f32 atomics.
// ---------------------------------------------------------------------------
__global__ __launch_bounds__(256) void gcn_edge_scatter(
    const int* __restrict__ src, const int* __restrict__ dst,
    const float* __restrict__ dinv, const float* __restrict__ h,
    float* __restrict__ out, int E) {
    int e = blockIdx.x * 4 + (threadIdx.x >> 6);
    if (e >= E) return;
    int f = (threadIdx.x & 63) * 4;
    int s = src[e], d = dst[e];
    float nrm = dinv[s] * dinv[d];
    const float4 v = *(const float4*)(h + (size_t)s * DH + f);
    float* o = out + (size_t)d * DH + f;
    atomicAdd(o + 0, v.x * nrm);
    atomicAdd(o + 1, v.y * nrm);
    atomicAdd(o + 2, v.z * nrm);
    atomicAdd(o + 3, v.w * nrm);
}

// ---------------------------------------------------------------------------
// LayerNorm + ReLU, one wave32 per node row (8 floats/lane), in place.
// ---------------------------------------------------------------------------
__global__ __launch_bounds__(256) void gcn_ln_relu(
    float* __restrict__ x, const float* __restrict__ w,
    const float* __restrict__ b, int N) {
    int wave = threadIdx.x >> 5;
    int lane = threadIdx.x & 31;
    int row  = blockIdx.x * 8 + wave;
    if (row >= N) return;
    float* r = x + (size_t)row * DH + lane * 8;

    float4 v0 = *(const float4*)(r);
    float4 v1 = *(const float4*)(r + 4);
    float vals[8] = {v0.x, v0.y, v0.z, v0.w, v1.x, v1.y, v1.z, v1.w};

    float s = 0.f, ss = 0.f;
    #pragma unroll
    for (int j = 0; j < 8; ++j) { s += vals[j]; ss += vals[j] * vals[j]; }
    #pragma unroll
    for (int m = 16; m > 0; m >>= 1) {   // wave32 butterfly
        s  += __shfl_xor(s,  m, 32);
        ss += __shfl_xor(ss, m, 32);
    }
    float mu  = s * (1.0f / DH);
    float var = ss * (1.0f / DH) - mu * mu;
    float inv = rsqrtf(var + 1e-5f);

    #pragma unroll
    for (int j = 0; j < 8; ++j) {
        int f = lane * 8 + j;
        float y = (vals[j] - mu) * inv * w[f] + b[f];
        r[j] = y > 0.f ? y : 0.f;
    }
}

// ---------------------------------------------------------------------------
// Pooling: zero, scatter-sum per graph (+count), then mean+linear head.
// ---------------------------------------------------------------------------
__global__ void gcn_zero(float* __restrict__ p, int n) {
    int i = blockIdx.x * blockDim.x + threadIdx.x;
    if (i < n) p[i] = 0.0f;
}

__global__ __launch_bounds__(256) void gcn_pool_scatter(
    const float* __restrict__ h, const int* __restrict__ batch,
    float* __restrict__ pooled, float* __restrict__ cnt) {
    int i = blockIdx.x;
    int f = threadIdx.x;
    int g = batch[i];
    atomicAdd(pooled + (size_t)g * DH + f, h[(size_t)i * DH + f]);
    if (f == 0) atomicAdd(cnt + g, 1.0f);
}

__global__ __launch_bounds__(32) void gcn_head(
    const float* __restrict__ pooled, const float* __restrict__ cnt,
    const float* __restrict__ W, const float* __restrict__ b,
    float* __restrict__ out, int C) {
    int g = blockIdx.x;
    int c = threadIdx.x;
    if (c >= C) return;
    float cn = cnt[g];
    float inv = 1.0f / (cn > 1.0f ? cn : 1.0f);
    float acc = b[c];
    for (int k = 0; k < DH; ++k)
        acc += pooled[(size_t)g * DH + k] * inv * W[(size_t)k * C + c];
    out[(size_t)g * C + c] = acc;
}

// ---------------------------------------------------------------------------
// Host driver
// ---------------------------------------------------------------------------
extern "C" void kernel_launch(void* const* d_in, const int* in_sizes, int n_in,
                              void* d_out, int out_size, void* d_ws, size_t ws_size,
                              hipStream_t stream) {
    const float* x     = (const float*)d_in[0];
    const int*   ei    = (const int*)  d_in[1];
    const int*   batch = (const int*)  d_in[2];
    const float* W1    = (const float*)d_in[3];
    const float* b1    = (const float*)d_in[4];
    const float* W2    = (const float*)d_in[5];
    const float* b2    = (const float*)d_in[6];
    const float* ln1w  = (const float*)d_in[7];
    const float* ln1b  = (const float*)d_in[8];
    const float* ln2w  = (const float*)d_in[9];
    const float* ln2b  = (const float*)d_in[10];
    const float* linW  = (const float*)d_in[11];
    const float* linb  = (const float*)d_in[12];
    float*       out   = (float*)d_out;

    const int N = in_sizes[0] / DH;     // 50000
    const int E = in_sizes[1] / 2;      // 800000
    const int C = 10;
    const int G = out_size / C;         // 512

    const int* srcI = ei;
    const int* dstI = ei + E;

    // workspace carving
    float* bufA   = (float*)d_ws;                  // GEMM outputs   [N,256]
    float* bufB   = bufA + (size_t)N * DH;         // aggregated h   [N,256]
    float* dinv   = bufB + (size_t)N * DH;         // [N]
    float* pooled = dinv + N;                      // [G,256]
    float* cnt    = pooled + (size_t)G * DH;       // [G] (contiguous after pooled)

    // symmetric normalization (shared by both layers)
    gcn_deg_init <<<(N + 255) / 256, 256, 0, stream>>>(dinv, N);
    gcn_deg_count<<<(E + 255) / 256, 256, 0, stream>>>(dstI, dinv, E);
    gcn_deg_rsqrt<<<(N + 255) / 256, 256, 0, stream>>>(dinv, N);

    // ---- layer 1 ----
    gcn_gemm_wmma  <<<N / 16, 128, 0, stream>>>(x, W1, bufA, N);
    gcn_agg_init   <<<N, DH, 0, stream>>>(bufA, dinv, b1, bufB);
    gcn_edge_scatter<<<(E + 3) / 4, 256, 0, stream>>>(srcI, dstI, dinv, bufA, bufB, E);
    gcn_ln_relu    <<<(N + 7) / 8, 256, 0, stream>>>(bufB, ln1w, ln1b, N);

    // ---- layer 2 ----
    gcn_gemm_wmma  <<<N / 16, 128, 0, stream>>>(bufB, W2, bufA, N);
    gcn_agg_init   <<<N, DH, 0, stream>>>(bufA, dinv, b2, bufB);
    gcn_edge_scatter<<<(E + 3) / 4, 256, 0, stream>>>(srcI, dstI, dinv, bufA, bufB, E);
    gcn_ln_relu    <<<(N + 7) / 8, 256, 0, stream>>>(bufB, ln2w, ln2b, N);

    // ---- pool + classifier head ----
    int poolElems = G * DH + G;                    // pooled + cnt, contiguous
    gcn_zero        <<<(poolElems + 255) / 256, 256, 0, stream>>>(pooled, poolElems);
    gcn_pool_scatter<<<N, DH, 0, stream>>>(bufB, batch, pooled, cnt);
    gcn_head        <<<G, 32, 0, stream>>>(pooled, cnt, linW, linb, out, C);
}